// TimeLSTM_module_14319420965205
// MI455X (gfx1250) — compile-verified
//
#include <hip/hip_runtime.h>
#include <math.h>

#define BB 64
#define SS 1024
#define IND 256
#define HD 512
#define GD (4 * HD)      // 2048
#define NBLOCKS 32       // 32 blocks x 4 waves = 128 waves = 4 M-tiles x 32 J-tiles

typedef float v2f __attribute__((ext_vector_type(2)));
typedef float v8f __attribute__((ext_vector_type(8)));

__device__ __forceinline__ v8f wmma_f32x4(v2f a, v2f b, v8f c) {
    // D = A(16x4 f32) * B(4x16 f32) + C(16x16 f32), fp32 matrix pipe
    return __builtin_amdgcn_wmma_f32_16x16x4_f32(false, a, false, b, (short)0, c, false, false);
}

__device__ __forceinline__ float sigm(float x) { return 1.0f / (1.0f + __expf(-x)); }

// Device-wide split barrier (32 resident blocks). bar[0]=count, bar[1]=generation.
__device__ __forceinline__ void grid_sync(int* bar) {
    __threadfence();                 // release: this block's h/c stores reach L2
    __syncthreads();
    if (threadIdx.x == 0) {
        int g = atomicAdd(&bar[1], 0);
        if (atomicAdd(&bar[0], 1) == NBLOCKS - 1) {
            atomicExch(&bar[0], 0);
            atomicAdd(&bar[1], 1);
        } else {
            while (atomicAdd(&bar[1], 0) == g) { __builtin_amdgcn_s_sleep(4); }
        }
    }
    __syncthreads();
    __threadfence();                 // acquire: drop stale WGP-cache lines before reading new h/c
}

__global__ void init_ws(float* __restrict__ h0, float* __restrict__ c0, int* __restrict__ bar) {
    int i = blockIdx.x * blockDim.x + threadIdx.x;
    if (i < BB * HD) { h0[i] = 0.0f; c0[i] = 0.0f; }
    if (i == 0) { bar[0] = 0; bar[1] = 0; }
}

// Persistent scan kernel: one launch runs all 1024 time steps.
// Wave tile: rows m0..m0+15 (batch), cols j0..j0+15 of EACH gate (f,i,o,ct) + of c@W_d.
// 5 fp32 WMMA accumulators per wave -> full cell update fuses in registers.
__global__ __launch_bounds__(128)
void lstm_scan(const float* __restrict__ x,      // [B,S,IN]
               const float* __restrict__ ts,     // [B,S]
               const int*   __restrict__ lens,   // [B]
               const float* __restrict__ W_all,  // [H,4H]
               const float* __restrict__ b_all,  // [4H]
               const float* __restrict__ U_all,  // [IN,4H]
               const float* __restrict__ b_u,    // [4H]
               const float* __restrict__ W_d,    // [H,H]
               const float* __restrict__ b_d,    // [H]
               float* __restrict__ out,          // outputs | last_h | last_c
               float* __restrict__ h0, float* __restrict__ h1,
               float* __restrict__ c0, float* __restrict__ c1,
               int* __restrict__ bar)
{
    const int lane = threadIdx.x & 31;
    const int gwid = blockIdx.x * 4 + (threadIdx.x >> 5);   // 0..127
    const int m0   = (gwid & 3) * 16;                       // batch tile
    const int j0   = (gwid >> 2) * 16;                      // hidden-column tile
    const int r    = lane & 15;
    const int kb   = (lane >> 4) << 1;                      // K pair: lanes 0-15 -> +0,+1 ; 16-31 -> +2,+3
    const int j    = j0 + r;

    // per-lane constants (fixed tile ownership across all steps)
    const float bias_f = b_all[j]            + b_u[j];
    const float bias_i = b_all[HD + j]       + b_u[HD + j];
    const float bias_o = b_all[2 * HD + j]   + b_u[2 * HD + j];
    const float bias_t = b_all[3 * HD + j]   + b_u[3 * HD + j];
    const float bias_d = b_d[j];

    const int rowBase = m0 + ((lane >> 4) << 3);            // C/D layout: row = rowBase + v
    int lenv[8];
#pragma unroll
    for (int v = 0; v < 8; ++v) lenv[v] = lens[rowBase + v];

    float creg[8];                                          // register-resident c state for owned cells
#pragma unroll
    for (int v = 0; v < 8; ++v) creg[v] = 0.0f;

    float* hr = h0; float* cr = c0;                         // read  (step t state)
    float* hw = h1; float* cw = c1;                         // write (step t+1 state)

    float* last_h = out + (size_t)BB * SS * HD;
    float* last_c = last_h + (size_t)BB * HD;

    const int arow = (m0 + r) * HD;                         // A-fragment row offset into h/c

    for (int t = 0; t < SS; ++t) {
        v8f af = {}, ai = {}, ao = {}, act = {}, ad = {};

        // ---- phase 1: K in [0,512): h @ W_all (4 gate cols) and c @ W_d ----
#pragma unroll 2
        for (int kk = 0; kk < HD; kk += 4) {
            const int ka = kk + kb;
            v2f a_h = *(const v2f*)(hr + arow + ka);
            v2f a_c = *(const v2f*)(cr + arow + ka);
            const float* Bw = W_all + (size_t)ka * GD;
            const float* Bd = W_d   + (size_t)ka * HD;
            v2f bf = { Bw[j],            Bw[GD + j] };
            v2f bi = { Bw[HD + j],       Bw[GD + HD + j] };
            v2f bo = { Bw[2 * HD + j],   Bw[GD + 2 * HD + j] };
            v2f bt = { Bw[3 * HD + j],   Bw[GD + 3 * HD + j] };
            v2f bd = { Bd[j],            Bd[HD + j] };
            af  = wmma_f32x4(a_h, bf, af);
            ai  = wmma_f32x4(a_h, bi, ai);
            ao  = wmma_f32x4(a_h, bo, ao);
            act = wmma_f32x4(a_h, bt, act);
            ad  = wmma_f32x4(a_c, bd, ad);
        }

        // ---- phase 2: K in [512,768): x_t @ U_all (4 gate cols) ----
        const float* xrow = x + ((size_t)(m0 + r) * SS + t) * IND;
#pragma unroll 2
        for (int kk = 0; kk < IND; kk += 4) {
            const int ka = kk + kb;
            v2f a = *(const v2f*)(xrow + ka);
            const float* Bu = U_all + (size_t)ka * GD;
            v2f bf = { Bu[j],            Bu[GD + j] };
            v2f bi = { Bu[HD + j],       Bu[GD + HD + j] };
            v2f bo = { Bu[2 * HD + j],   Bu[GD + 2 * HD + j] };
            v2f bt = { Bu[3 * HD + j],   Bu[GD + 3 * HD + j] };
            af  = wmma_f32x4(a, bf, af);
            ai  = wmma_f32x4(a, bi, ai);
            ao  = wmma_f32x4(a, bo, ao);
            act = wmma_f32x4(a, bt, act);
        }

        // ---- fused cell update, entirely in registers ----
#pragma unroll
        for (int v = 0; v < 8; ++v) {
            const int b = rowBase + v;
            const float g_t = 1.0f / logf(2.718281828459045f + ts[b * SS + t]);
            const float fv  = sigm(af[v]  + bias_f);
            const float iv  = sigm(ai[v]  + bias_i);
            const float ov  = sigm(ao[v]  + bias_o);
            const float ctv = tanhf(act[v] + bias_t);
            const float cs1 = tanhf(ad[v]  + bias_d);
            const float cold = creg[v];
            const float cadj = cold - cs1 + cs1 * g_t;
            const float cn = fv * cadj + iv * ctv;
            const float hn = ov * tanhf(cn);
            creg[v] = cn;
            hw[b * HD + j] = hn;
            cw[b * HD + j] = cn;
            out[((size_t)b * SS + t) * HD + j] = (t < lenv[v]) ? hn : 0.0f;
            if (t == lenv[v] - 1) {
                last_h[b * HD + j] = hn;
                last_c[b * HD + j] = cn;
            }
        }

        // one device-wide barrier per step; ping-pong swap
        grid_sync(bar);
        float* tmp;
        tmp = hr; hr = hw; hw = tmp;
        tmp = cr; cr = cw; cw = tmp;
    }
}

extern "C" void kernel_launch(void* const* d_in, const int* in_sizes, int n_in,
                              void* d_out, int out_size, void* d_ws, size_t ws_size,
                              hipStream_t stream) {
    const float* inputs = (const float*)d_in[0];
    const float* ts     = (const float*)d_in[1];
    const int*   lens   = (const int*)d_in[2];
    const float* W_all  = (const float*)d_in[3];
    const float* b_all  = (const float*)d_in[4];
    const float* U_all  = (const float*)d_in[5];
    const float* b_u    = (const float*)d_in[6];
    const float* W_d    = (const float*)d_in[7];
    const float* b_d    = (const float*)d_in[8];
    float* out = (float*)d_out;

    // workspace: h0 | h1 | c0 | c1 (each B*H floats) | barrier (2 ints)  ~512 KB
    float* h0 = (float*)d_ws;
    float* h1 = h0 + (size_t)BB * HD;
    float* c0 = h1 + (size_t)BB * HD;
    float* c1 = c0 + (size_t)BB * HD;
    int*   bar = (int*)(c1 + (size_t)BB * HD);

    hipLaunchKernelGGL(init_ws, dim3((BB * HD + 255) / 256), dim3(256), 0, stream, h0, c0, bar);
    hipLaunchKernelGGL(lstm_scan, dim3(NBLOCKS), dim3(128), 0, stream,
                       inputs, ts, lens, W_all, b_all, U_all, b_u, W_d, b_d,
                       out, h0, h1, c0, c1, bar);
}